// GCN_43585328119844
// MI455X (gfx1250) — compile-verified
//
#include <hip/hip_runtime.h>
#include <hip/hip_bf16.h>

typedef __attribute__((ext_vector_type(2))) float v2f;
typedef __attribute__((ext_vector_type(8))) float v8f;

#define THREADS 256

// ---------------- degree counting ----------------
__global__ void GCN_count_deg(const int* __restrict__ src, const int* __restrict__ dst,
                              unsigned* __restrict__ cnt_out, unsigned* __restrict__ cnt_in, int E) {
    int e = blockIdx.x * blockDim.x + threadIdx.x;
    if (e < E) {
        atomicAdd(&cnt_out[src[e]], 1u);
        atomicAdd(&cnt_in[dst[e]], 1u);
    }
}

// ---------------- rsqrt(max(deg,1)) ----------------
__global__ void GCN_rsqrt(const unsigned* __restrict__ co, const unsigned* __restrict__ ci,
                          float* __restrict__ ro, float* __restrict__ ri, int N) {
    int n = blockIdx.x * blockDim.x + threadIdx.x;
    if (n < N) {
        ro[n] = rsqrtf(fmaxf((float)co[n], 1.0f));
        ri[n] = rsqrtf(fmaxf((float)ci[n], 1.0f));
    }
}

// ---------------- (B,h,T,N) -> node-major (N,B,T,h), scaled by rsqrt(deg_out) ----------------
__global__ void GCN_permute_scale(const float* __restrict__ in_feat, const float* __restrict__ rs_out,
                                  float* __restrict__ xhat, int N) {
    __shared__ float tile[64 * 33];  // pad stride 33 to dodge bank conflicts
    int bt = blockIdx.y;             // bt = b*4 + t
    int b = bt >> 2, t = bt & 3;
    int n0 = blockIdx.x * 32;
    for (int i = threadIdx.x; i < 64 * 32; i += THREADS) {
        int nl = i & 31, h = i >> 5;
        int n = n0 + nl;
        float v = 0.0f;
        if (n < N) v = in_feat[(size_t)((b * 64 + h) * 4 + t) * N + n];
        tile[h * 33 + nl] = v;
    }
    __syncthreads();
    for (int i = threadIdx.x; i < 64 * 32; i += THREADS) {
        int h = i & 63, nl = i >> 6;
        int n = n0 + nl;
        if (n < N)
            xhat[(size_t)((n * 2 + b) * 4 + t) * 64 + h] = tile[h * 33 + nl] * rs_out[n];
    }
}

// ---------------- edge scatter: agg[dst] += xhat[src] ----------------
// One wave per edge: 4 x float4 coalesced reads (node vectors are L2-resident),
// 16 global_atomic_add_f32 per lane into the L2-resident agg buffer.
__global__ void GCN_scatter(const int* __restrict__ src, const int* __restrict__ dst,
                            const float* __restrict__ xhat, float* __restrict__ agg, int E) {
    int gwave = (int)((blockIdx.x * (unsigned)blockDim.x + threadIdx.x) >> 5);
    int lane = threadIdx.x & 31;
    if (gwave >= E) return;
    int s = src[gwave], d = dst[gwave];
    const float4* xs = (const float4*)(xhat + (size_t)s * 512);
    float* ad = agg + (size_t)d * 512;
#pragma unroll
    for (int i = 0; i < 4; ++i) {
        float4 v = xs[lane + 32 * i];
        int base = (lane + 32 * i) * 4;
        atomicAdd(ad + base + 0, v.x);
        atomicAdd(ad + base + 1, v.y);
        atomicAdd(ad + base + 2, v.z);
        atomicAdd(ad + base + 3, v.w);
    }
}

// ---------------- per-node-pair GEMM via fp32 WMMA, fused scale/bias/relu/transpose ----------------
// Wave handles nodes (n0, n0+1): A = contiguous 16x64 fp32 block of agg (rows
// scaled by rsqrt(deg_in)), B = W staged K-pair-interleaved in LDS via gfx1250
// async-to-LDS loads, D = 16x64 via 4 column tiles x 16 steps of
// V_WMMA_F32_16X16X4_F32.
__global__ void __launch_bounds__(THREADS)
GCN_gemm_relu(const float* __restrict__ agg, const float* __restrict__ rs_in,
              const float* __restrict__ W, const float* __restrict__ bias,
              float* __restrict__ out, int N) {
    // Interleaved layout: WlB[(k>>1)*160 + 2*f + (k&1)] = W[k*64+f].
    // Row stride 160 dwords => odd/even K rows are 32 banks apart, so the two
    // lane-halves of a wave (which read different K rows) never collide.
    __shared__ float WlB[32 * 160];
    __shared__ float bl[64];

    // Async staging of W into LDS (per-lane LDS addresses do the interleave).
#pragma unroll
    for (int it = 0; it < 16; ++it) {
        int i = it * THREADS + threadIdx.x;
        int k = i >> 6, f = i & 63;
        unsigned ldsaddr = (unsigned)(size_t)&WlB[(k >> 1) * 160 + 2 * f + (k & 1)];
        unsigned long long gaddr = (unsigned long long)(size_t)(W + i);
        asm volatile("global_load_async_to_lds_b32 %0, %1, off"
                     :: "v"(ldsaddr), "v"(gaddr) : "memory");
    }
    if (threadIdx.x < 64) bl[threadIdx.x] = bias[threadIdx.x];
    asm volatile("s_wait_asynccnt 0x0" ::: "memory");
    __syncthreads();

    int wave = threadIdx.x >> 5;
    int lane = threadIdx.x & 31;
    int pair = blockIdx.x * 8 + wave;   // 8 waves per block, 1 node-pair per wave
    int n0 = pair * 2;
    if (n0 + 1 >= N) return;            // wave-uniform: EXEC stays all-ones for WMMA

    int m = lane & 15;                  // A-matrix row held by this lane
    int koff = (lane >> 4) * 2;         // lane half selects K pair {0,1} / {2,3}
    int fl = lane & 15;                 // B-matrix column held by this lane
    float rs = rs_in[n0 + (m >> 3)];    // rsqrt(deg_in) of the row's node

    // Preload this lane's entire A fragment: 16 independent 8B loads pipeline
    // through LOADcnt instead of a load->wait->wmma chain per K-step.
    const float* A = agg + (size_t)n0 * 512 + m * 64 + koff;
    v2f araw[16];
#pragma unroll
    for (int kk = 0; kk < 16; ++kk) araw[kk] = *(const v2f*)(A + kk * 4);

    v8f acc[4] = {};                    // 4 column tiles of 16 cols each
#pragma unroll
    for (int kk = 0; kk < 16; ++kk) {
        v2f a = araw[kk] * rs;          // v_pk_mul_f32 row scaling
        int row = 2 * kk + (koff >> 1);
        const v2f* Bp = (const v2f*)(WlB + row * 160 + 2 * fl);
        v2f b0 = Bp[0];                 // single ds_load_b64 per operand,
        v2f b1 = Bp[16];                // no repacking movs
        v2f b2 = Bp[32];
        v2f b3 = Bp[48];
        acc[0] = __builtin_amdgcn_wmma_f32_16x16x4_f32(false, a, false, b0, (short)0, acc[0], false, false);
        acc[1] = __builtin_amdgcn_wmma_f32_16x16x4_f32(false, a, false, b1, (short)0, acc[1], false, false);
        acc[2] = __builtin_amdgcn_wmma_f32_16x16x4_f32(false, a, false, b2, (short)0, acc[2], false, false);
        acc[3] = __builtin_amdgcn_wmma_f32_16x16x4_f32(false, a, false, b3, (short)0, acc[3], false, false);
    }

    // D layout: VGPR j, lanes 0-15 -> M=j, lanes 16-31 -> M=j+8; N = lane&15.
    int mh = (lane >> 4) * 8;
#pragma unroll
    for (int c = 0; c < 4; ++c) {
        int f = c * 16 + fl;
        float bv = bl[f];
#pragma unroll
        for (int j = 0; j < 8; ++j) {
            float v = fmaxf(acc[c][j] + bv, 0.0f);
            int M = j + mh;
            int nl = M >> 3, r = M & 7, bb = r >> 2, tt = r & 3;
            out[(size_t)((bb * 64 + f) * 4 + tt) * N + (n0 + nl)] = v;
        }
    }
}

extern "C" void kernel_launch(void* const* d_in, const int* in_sizes, int n_in,
                              void* d_out, int out_size, void* d_ws, size_t ws_size,
                              hipStream_t stream) {
    const float* in_feat = (const float*)d_in[0];
    const int*   src     = (const int*)d_in[1];
    const int*   dst     = (const int*)d_in[2];
    const float* W       = (const float*)d_in[3];
    const float* bias    = (const float*)d_in[4];
    float* out = (float*)d_out;

    int N = in_sizes[0] / 512;   // 2 * 64 * 4 features per node
    int E = in_sizes[1];

    char* ws = (char*)d_ws;
    size_t featBytes = (size_t)N * 512 * sizeof(float);
    float*    xhat    = (float*)ws;
    float*    agg     = (float*)(ws + featBytes);
    unsigned* cnt_out = (unsigned*)(ws + 2 * featBytes);
    unsigned* cnt_in  = cnt_out + N;
    float*    rs_out  = (float*)(cnt_in + N);
    float*    rs_in   = rs_out + N;

    hipMemsetAsync(agg, 0, featBytes, stream);
    hipMemsetAsync(cnt_out, 0, (size_t)2 * N * sizeof(unsigned), stream);

    GCN_count_deg<<<(E + THREADS - 1) / THREADS, THREADS, 0, stream>>>(src, dst, cnt_out, cnt_in, E);
    GCN_rsqrt<<<(N + THREADS - 1) / THREADS, THREADS, 0, stream>>>(cnt_out, cnt_in, rs_out, rs_in, N);

    dim3 pg((N + 31) / 32, 8);
    GCN_permute_scale<<<pg, THREADS, 0, stream>>>(in_feat, rs_out, xhat, N);

    int scatterBlocks = (int)(((long long)E * 32 + THREADS - 1) / THREADS);
    GCN_scatter<<<scatterBlocks, THREADS, 0, stream>>>(src, dst, xhat, agg, E);

    int pairs = N / 2;
    GCN_gemm_relu<<<(pairs + 7) / 8, THREADS, 0, stream>>>(agg, rs_in, W, bias, out, N);
}